// GCN_62371515072934
// MI455X (gfx1250) — compile-verified
//
#include <hip/hip_runtime.h>

typedef float v2f __attribute__((ext_vector_type(2)));
typedef float v8f __attribute__((ext_vector_type(8)));

#define NFEAT 64

// ---------------------------------------------------------------------------
// Small utility kernels
// ---------------------------------------------------------------------------
__global__ void gcn_zero_f32(float* __restrict__ p, int count) {
  int i = blockIdx.x * blockDim.x + threadIdx.x;
  if (i < count) p[i] = 0.0f;
}

__global__ void gcn_deg_init(float* __restrict__ deg, int n) {
  int i = blockIdx.x * blockDim.x + threadIdx.x;
  if (i < n) deg[i] = 1.0f;  // self loop
}

__global__ void gcn_deg_count(const int* __restrict__ dst, float* __restrict__ deg, int E) {
  int i = blockIdx.x * blockDim.x + threadIdx.x;
  if (i < E) atomicAdd(&deg[dst[i]], 1.0f);
}

__global__ void gcn_deg_rsqrt(float* __restrict__ deg, int n) {
  int i = blockIdx.x * blockDim.x + threadIdx.x;
  if (i < n) deg[i] = rsqrtf(deg[i]);  // deg >= 1 always (self loop)
}

// ---------------------------------------------------------------------------
// GEMM: out[M,64] = in[M,K] @ W[K,64], fp32 via V_WMMA_F32_16X16X4_F32.
// M multiple of 32 (100000 = 32*3125), K in {128,64}. Block = 256 thr = 8 waves.
// Each block: 32 rows x 64 cols; wave w -> rowTile = w>>2, colTile = w&3.
// A tile staged in LDS (row pad +4 floats to avoid bank conflicts).
// ---------------------------------------------------------------------------
__global__ __launch_bounds__(256) void gcn_gemm_wmma_f32(
    const float* __restrict__ in, const float* __restrict__ W,
    float* __restrict__ out, int K) {
  __shared__ float As[32 * (128 + 4)];  // max K = 128
  const int ldA = K + 4;
  const int tid = threadIdx.x;
  const int blockRow0 = blockIdx.x * 32;

  // cooperative coalesced load of the 32 x K A-tile
  const int total = 32 * K;
  for (int idx = tid; idx < total; idx += 256) {
    int r = idx / K;
    int c = idx - r * K;
    As[r * ldA + c] = in[(size_t)(blockRow0 + r) * K + c];
  }
  __syncthreads();

  const int wave = tid >> 5;
  const int lane = tid & 31;
  const int m  = lane & 15;       // row within tile (A) / col within tile (B,C)
  const int kp = lane >> 4;       // lane-half selector
  const int rowL = (wave >> 2) * 16;   // local row tile: 0 or 16
  const int col0 = (wave & 3) * 16;    // 0,16,32,48

  v8f c = {};
  for (int kk = 0; kk < K; kk += 4) {
    const int ka = kk + 2 * kp;   // K index for VGPR j: ka + j
    v2f a, b;
    a.x = As[(rowL + m) * ldA + ka];
    a.y = As[(rowL + m) * ldA + ka + 1];
    b.x = W[(size_t)(ka + 0) * NFEAT + col0 + m];
    b.y = W[(size_t)(ka + 1) * NFEAT + col0 + m];
    c = __builtin_amdgcn_wmma_f32_16x16x4_f32(
        /*neg_a=*/false, a, /*neg_b=*/false, b,
        /*c_mod=*/(short)0, c, /*reuse_a=*/false, /*reuse_b=*/false);
  }

  // C/D layout: VGPR i holds row (i + 8*kp), cols = lanes (m)
  const int gRow0 = blockRow0 + rowL;
#pragma unroll
  for (int i = 0; i < 8; ++i) {
    out[(size_t)(gRow0 + i + 8 * kp) * NFEAT + col0 + m] = c[i];
  }
}

// ---------------------------------------------------------------------------
// Edge scatter: acc[dst] += h[src] * inv_sqrt[src]*inv_sqrt[dst]
// 16 threads per edge, float4 per thread -> 64 feats, coalesced gather.
// ---------------------------------------------------------------------------
__global__ void gcn_scatter(const int* __restrict__ src, const int* __restrict__ dst,
                            const float* __restrict__ h, const float* __restrict__ isq,
                            float* __restrict__ acc, int E) {
  int tid = blockIdx.x * blockDim.x + threadIdx.x;
  int e = tid >> 4;
  if (e >= E) return;
  int f = (tid & 15) * 4;
  int s = src[e], d = dst[e];
  float coef = isq[s] * isq[d];
  float4 hv = *(const float4*)(h + (size_t)s * NFEAT + f);
  float* o = acc + (size_t)d * NFEAT + f;
  atomicAdd(o + 0, hv.x * coef);
  atomicAdd(o + 1, hv.y * coef);
  atomicAdd(o + 2, hv.z * coef);
  atomicAdd(o + 3, hv.w * coef);
}

// ---------------------------------------------------------------------------
// Finalize a GCN layer: out = relu(acc + h * inv_sqrt^2 (self loop) + bias)
// Elementwise; acc/out may alias.
// ---------------------------------------------------------------------------
__global__ void gcn_finalize(const float* __restrict__ acc, const float* __restrict__ h,
                             const float* __restrict__ isq, const float* __restrict__ bias,
                             float* __restrict__ out, int n) {
  int t = blockIdx.x * blockDim.x + threadIdx.x;
  int i = t >> 6;
  if (i >= n) return;
  int f = t & 63;
  float is = isq[i];
  float v = acc[(size_t)i * NFEAT + f] + h[(size_t)i * NFEAT + f] * is * is + bias[f];
  out[(size_t)i * NFEAT + f] = fmaxf(v, 0.0f);
}

// ---------------------------------------------------------------------------
// Segment pool: sums + counts per graph.
// ---------------------------------------------------------------------------
__global__ void gcn_pool(const float* __restrict__ h, const int* __restrict__ batch,
                         float* __restrict__ gsum, float* __restrict__ gcnt, int n) {
  int t = blockIdx.x * blockDim.x + threadIdx.x;
  int i = t >> 6;
  if (i >= n) return;
  int f = t & 63;
  int g = batch[i];
  atomicAdd(&gsum[g * NFEAT + f], h[(size_t)i * NFEAT + f]);
  if (f == 0) atomicAdd(&gcnt[g], 1.0f);
}

// ---------------------------------------------------------------------------
// Head: g = gsum/max(cnt,1); z = g@Wp + bp; logits = z@Wc + bc.
// Single block; sizes are tiny (64x64x64 + 64x64x6).
// out layout: [0..383] = logits (64x6), [384..4479] = z (64x64).
// ---------------------------------------------------------------------------
__global__ __launch_bounds__(256) void gcn_head(
    const float* __restrict__ gsum, const float* __restrict__ gcnt,
    const float* __restrict__ Wp, const float* __restrict__ bp,
    const float* __restrict__ Wc, const float* __restrict__ bc,
    float* __restrict__ out) {
  __shared__ float g[64 * 64];
  __shared__ float zs[64 * 64];
  const int t = threadIdx.x;
  for (int idx = t; idx < 64 * 64; idx += 256) {
    int gi = idx >> 6;
    float c = fmaxf(gcnt[gi], 1.0f);
    g[idx] = gsum[idx] / c;
  }
  __syncthreads();
  for (int idx = t; idx < 64 * 64; idx += 256) {
    int m = idx >> 6, nn = idx & 63;
    float a = bp[nn];
#pragma unroll 8
    for (int k = 0; k < 64; ++k) a += g[m * 64 + k] * Wp[k * 64 + nn];
    zs[idx] = a;
    out[384 + idx] = a;  // z
  }
  __syncthreads();
  for (int idx = t; idx < 64 * 6; idx += 256) {
    int m = idx / 6, nn = idx - m * 6;
    float a = bc[nn];
#pragma unroll 8
    for (int k = 0; k < 64; ++k) a += zs[m * 64 + k] * Wc[k * 6 + nn];
    out[idx] = a;  // logits
  }
}

// ---------------------------------------------------------------------------
extern "C" void kernel_launch(void* const* d_in, const int* in_sizes, int n_in,
                              void* d_out, int out_size, void* d_ws, size_t ws_size,
                              hipStream_t stream) {
  const float* x     = (const float*)d_in[0];
  const int*   ei    = (const int*)d_in[1];
  const int*   batch = (const int*)d_in[2];
  const float* W1    = (const float*)d_in[3];
  const float* b1    = (const float*)d_in[4];
  const float* W2    = (const float*)d_in[5];
  const float* b2    = (const float*)d_in[6];
  const float* Wp    = (const float*)d_in[7];
  const float* bp    = (const float*)d_in[8];
  const float* Wc    = (const float*)d_in[9];
  const float* bc    = (const float*)d_in[10];

  const int n = in_sizes[0] / 128;   // 100000
  const int E = in_sizes[1] / 2;     // 1600000
  const int* src = ei;
  const int* dst = ei + E;

  // workspace carve-out (256B aligned)
  char* ws = (char*)d_ws;
  size_t off = 0;
  auto carve = [&](size_t bytes) -> float* {
    float* p = (float*)(ws + off);
    off = (off + bytes + 255) & ~(size_t)255;
    return p;
  };
  float* isq  = carve((size_t)n * sizeof(float));           // inv sqrt degree
  float* bufA = carve((size_t)n * NFEAT * sizeof(float));   // GEMM output (h = x@W)
  float* bufB = carve((size_t)n * NFEAT * sizeof(float));   // scatter acc / layer output
  float* gsum = carve(64 * NFEAT * sizeof(float));
  float* gcnt = carve(64 * sizeof(float));

  const int T = 256;
  const int nBlk  = (n + T - 1) / T;
  const int eBlk  = (E + T - 1) / T;
  const int nfTot = n * NFEAT;
  const int nfBlk = (nfTot + T - 1) / T;
  const int e16Blk = (int)(((long long)E * 16 + T - 1) / T);
  const int gemmBlk = n / 32;  // 100000 % 32 == 0

  // degrees -> inv sqrt
  gcn_deg_init<<<nBlk, T, 0, stream>>>(isq, n);
  gcn_deg_count<<<eBlk, T, 0, stream>>>(dst, isq, E);
  gcn_deg_rsqrt<<<nBlk, T, 0, stream>>>(isq, n);

  // ---- layer 1 ----
  gcn_gemm_wmma_f32<<<gemmBlk, T, 0, stream>>>(x, W1, bufA, 128);
  gcn_zero_f32<<<nfBlk, T, 0, stream>>>(bufB, nfTot);
  gcn_scatter<<<e16Blk, T, 0, stream>>>(src, dst, bufA, isq, bufB, E);
  gcn_finalize<<<nfBlk, T, 0, stream>>>(bufB, bufA, isq, b1, bufB, n);  // h1 in bufB

  // ---- layer 2 ----
  gcn_gemm_wmma_f32<<<gemmBlk, T, 0, stream>>>(bufB, W2, bufA, 64);
  gcn_zero_f32<<<nfBlk, T, 0, stream>>>(bufB, nfTot);
  gcn_scatter<<<e16Blk, T, 0, stream>>>(src, dst, bufA, isq, bufB, E);
  gcn_finalize<<<nfBlk, T, 0, stream>>>(bufB, bufA, isq, b2, bufB, n);  // h2 in bufB

  // ---- pool + head ----
  gcn_zero_f32<<<(64 * NFEAT + 64 + T - 1) / T, T, 0, stream>>>(gsum, 64 * NFEAT + 64);
  gcn_pool<<<nfBlk, T, 0, stream>>>(bufB, batch, gsum, gcnt, n);
  gcn_head<<<1, T, 0, stream>>>(gsum, gcnt, Wp, bp, Wc, bc, (float*)d_out);
}